// TnoFft_66554813219027
// MI455X (gfx1250) — compile-verified
//
#include <hip/hip_runtime.h>

// ---------------------------------------------------------------------------
// TNO long-conv via fused FFT on gfx1250.
//   fwd:  radix-16 head stage as V_WMMA_F32_16X16X4_F32 complex GEMM,
//         twiddle, then 16 x 512-pt radix-2 DIF row FFTs (bit-rev output).
//   mul:  pointwise in the (consistently) scrambled spectral order.
//   inv:  512-pt radix-2 DIT rows w/ conjugate twiddles, conj twiddle,
//         conj WMMA radix-16 tail stage with 1/8192 scale.
//   I/O:  per-channel stride-512 columns moved by the Tensor Data Mover
//         (tensor_load_to_lds / tensor_store_from_lds), absorbed by 192MB L2.
// ---------------------------------------------------------------------------

typedef float v2f  __attribute__((ext_vector_type(2)));
typedef float v8f  __attribute__((ext_vector_type(8)));
typedef unsigned int u32x4 __attribute__((ext_vector_type(4)));
typedef int   i32x4 __attribute__((ext_vector_type(4)));
typedef int   i32x8 __attribute__((ext_vector_type(8)));

#define THREADS 256
#define TWO_PI  6.28318530717958647692f
#define TW8192  (6.28318530717958647692f / 8192.0f)

// ---- LDS byte offset: generic LDS pointers carry the offset in addr[31:0]
__device__ __forceinline__ unsigned lds_byte_off(const void* p) {
  return (unsigned)(unsigned long long)p;
}

// ---- Tensor Data Mover: 2-D descriptor, tile = 1 x nrows, row stride = d ----
__device__ __forceinline__ void tdm_desc(u32x4& g0, i32x8& g1, unsigned ldsoff,
                                         unsigned long long ga, unsigned nrows,
                                         unsigned stride_elems) {
  g0[0] = 1u;                                            // count = 1 (valid D#)
  g0[1] = ldsoff;                                        // lds_addr
  g0[2] = (unsigned)(ga & 0xFFFFFFFFull);                // global_addr[31:0]
  g0[3] = (unsigned)((ga >> 32) & 0x01FFFFFFull) | (2u << 30); // ga[56:32]|type=2
  g1[0] = (int)(2u << 16);                               // data_size = 4B
  g1[1] = (int)(1u << 16);                               // tensor_dim0 = 1
  g1[2] = (int)((nrows & 0xFFFFu) << 16);                // tensor_dim1[15:0]
  g1[3] = (int)((nrows >> 16) & 0xFFFFu) | (int)(1u << 16); // dim1 hi | tile_dim0=1
  g1[4] = (int)(nrows & 0xFFFFu);                        // tile_dim1 = nrows
  g1[5] = (int)stride_elems;                             // tensor_dim0_stride lo
  g1[6] = 0;
  g1[7] = 0;
}

__device__ __forceinline__ void tdm_load_col(void* lds_dst, const float* gsrc,
                                             unsigned nrows, unsigned stride) {
  u32x4 g0; i32x8 g1;
  tdm_desc(g0, g1, lds_byte_off(lds_dst), (unsigned long long)gsrc, nrows, stride);
  i32x4 z4 = {0, 0, 0, 0};
  i32x8 z8 = {0, 0, 0, 0, 0, 0, 0, 0};
  __builtin_amdgcn_tensor_load_to_lds(g0, g1, z4, z4, z8, 0);
}

__device__ __forceinline__ void tdm_store_col(const void* lds_src, float* gdst,
                                              unsigned nrows, unsigned stride) {
  u32x4 g0; i32x8 g1;
  tdm_desc(g0, g1, lds_byte_off(lds_src), (unsigned long long)gdst, nrows, stride);
  i32x4 z4 = {0, 0, 0, 0};
  i32x8 z8 = {0, 0, 0, 0, 0, 0, 0, 0};
  __builtin_amdgcn_tensor_store_from_lds(g0, g1, z4, z4, z8, 0);
}

// ---- fp32 WMMA 16x16x4 wrapper ----
__device__ __forceinline__ v8f wmma4(v2f a, v2f b, v8f c) {
  return __builtin_amdgcn_wmma_f32_16x16x4_f32(false, a, false, b, (short)0, c,
                                               false, false);
}

// ---- radix-16 head/tail stage: G[k1][r] = sum_a F16[k1][a] * Z[a][r] -------
// dir=+1: forward (F16 = e^{-2pi i jk/16}), fused post-twiddle e^{-2pi i k1 r/8192}
// dir=-1: inverse (conj F16), fused pre-twiddle e^{+2pi i k1 r/8192}, scale
// A layout (16x4 fp32): lanes 0-15 M=lane, vgpr.x/.y = K0/K1; lanes 16-31 K2/K3.
// B layout (4x16):      lanes 0-15 rows K0(K2)/K1(K3) in .x/.y, N = lane&15.
// C/D layout: acc[v] -> (M = v + 8*(lane>>4), N = lane&15).
__device__ __forceinline__ void stage16(float* RE, float* IM, int tid, int dir,
                                        float scale) {
  const int lane = tid & 31;
  const int wave = tid >> 5;
  const int hi   = lane >> 4;
  const int l15  = lane & 15;
  const float sgn = (float)dir;

  v2f aC[4], aSp[4], aSn[4];
#pragma unroll
  for (int q = 0; q < 4; ++q) {
    const int k0 = q * 4 + hi * 2;
    float s0, c0, s1, c1;
    __sincosf((TWO_PI / 16.0f) * (float)(l15 * k0), &s0, &c0);
    __sincosf((TWO_PI / 16.0f) * (float)(l15 * (k0 + 1)), &s1, &c1);
    aC[q].x  = c0;         aC[q].y  = c1;
    aSp[q].x = sgn * s0;   aSp[q].y = sgn * s1;   // +sigma*S
    aSn[q].x = -sgn * s0;  aSn[q].y = -sgn * s1;  // -sigma*S
  }

#pragma unroll
  for (int tt = 0; tt < 4; ++tt) {
    const int col = (wave * 4 + tt) * 16 + l15;   // this wave owns 4 col tiles
    v2f br[4], bi[4];
#pragma unroll
    for (int q = 0; q < 4; ++q) {
      const int r0 = q * 4 + hi * 2;
      float xr0 = RE[r0 * 512 + col],       xi0 = IM[r0 * 512 + col];
      float xr1 = RE[(r0 + 1) * 512 + col], xi1 = IM[(r0 + 1) * 512 + col];
      if (dir < 0) {  // inverse: pre-twiddle e^{+2pi i k1 r / 8192}
        float s, c;
        __sincosf(TW8192 * (float)(r0 * col), &s, &c);
        float u = xr0 * c - xi0 * s, v = xr0 * s + xi0 * c;
        xr0 = u; xi0 = v;
        __sincosf(TW8192 * (float)((r0 + 1) * col), &s, &c);
        u = xr1 * c - xi1 * s; v = xr1 * s + xi1 * c;
        xr1 = u; xi1 = v;
      }
      br[q].x = xr0; br[q].y = xr1;
      bi[q].x = xi0; bi[q].y = xi1;
    }
    // complex GEMM: Gre = C@zr + s*S@zi ; Gim = C@zi - s*S@zr
    v8f accR = {0.f, 0.f, 0.f, 0.f, 0.f, 0.f, 0.f, 0.f};
    v8f accI = {0.f, 0.f, 0.f, 0.f, 0.f, 0.f, 0.f, 0.f};
#pragma unroll
    for (int q = 0; q < 4; ++q) accR = wmma4(aC[q],  br[q], accR);
#pragma unroll
    for (int q = 0; q < 4; ++q) accR = wmma4(aSp[q], bi[q], accR);
#pragma unroll
    for (int q = 0; q < 4; ++q) accI = wmma4(aC[q],  bi[q], accI);
#pragma unroll
    for (int q = 0; q < 4; ++q) accI = wmma4(aSn[q], br[q], accI);
#pragma unroll
    for (int v = 0; v < 8; ++v) {
      const int row = v + 8 * hi;
      float gr = accR[v], gi = accI[v];
      if (dir > 0) {  // forward: post-twiddle e^{-2pi i k1 r / 8192}
        float s, c;
        __sincosf(-TW8192 * (float)(row * col), &s, &c);
        float u = gr * c - gi * s;
        gi = gr * s + gi * c;
        gr = u;
      } else {
        gr *= scale; gi *= scale;
      }
      RE[row * 512 + col] = gr;
      IM[row * 512 + col] = gi;
    }
  }
}

// ---- 16 independent 512-pt row FFTs. dir=+1: DIF (nat->bitrev),
//      dir=-1: DIT with conjugate twiddles (bitrev->nat). W[j]=e^{-2pi i j/512}
__device__ __forceinline__ void rowfft(float* RE, float* IM, const float* WR,
                                       const float* WI, int tid, int dir) {
  if (dir > 0) {
    for (int s = 0; s < 9; ++s) {
      const int len = 512 >> s, half = 256 >> s;
      __syncthreads();
      for (int i = tid; i < 4096; i += THREADS) {
        const int row = i >> 8, k = i & 255;
        const int t = k & (half - 1), blk = k >> (8 - s);
        const int i0 = row * 512 + blk * len + t, i1 = i0 + half;
        float ur = RE[i0], ui = IM[i0], vr = RE[i1], vi = IM[i1];
        RE[i0] = ur + vr; IM[i0] = ui + vi;
        const float dr = ur - vr, di = ui - vi;
        const int w = t << s;
        const float wr = WR[w], wi = WI[w];
        RE[i1] = dr * wr - di * wi;
        IM[i1] = dr * wi + di * wr;
      }
    }
  } else {
    for (int s = 8; s >= 0; --s) {
      const int len = 512 >> s, half = 256 >> s;
      __syncthreads();
      for (int i = tid; i < 4096; i += THREADS) {
        const int row = i >> 8, k = i & 255;
        const int t = k & (half - 1), blk = k >> (8 - s);
        const int i0 = row * 512 + blk * len + t, i1 = i0 + half;
        float ur = RE[i0], ui = IM[i0], vr = RE[i1], vi = IM[i1];
        const int w = t << s;
        const float wr = WR[w], wi = -WI[w];          // conjugate
        const float tr = vr * wr - vi * wi, ti = vr * wi + vi * wr;
        RE[i0] = ur + tr; IM[i0] = ui + ti;
        RE[i1] = ur - tr; IM[i1] = ui - ti;
      }
    }
  }
}

#define LDS_FLOATS (8192 + 8192 + 256 + 256)   // RE, IM, WR, WI  (66 KiB)

// ---- kernel 1: That[c] = fwd_fft(pad(t[:,c])), planar per-channel in ws ----
__global__ void __launch_bounds__(THREADS)
tno_tfft_kernel(const float* __restrict__ t, float* __restrict__ ws) {
  extern __shared__ float lds[];
  float* RE = lds;
  float* IM = lds + 8192;
  float* WR = lds + 16384;
  float* WI = lds + 16640;
  const int tid = threadIdx.x;
  const int c = blockIdx.x;

  if (tid < 32) tdm_load_col(IM, t + c, 8190u, 512u);   // wave 0 drives the TDM
  {  // 512-pt twiddle table while the DMA runs
    float s, co;
    __sincosf(-TWO_PI * (float)tid / 512.0f, &s, &co);
    WR[tid] = co; WI[tid] = s;
  }
  if (tid < 32) __builtin_amdgcn_s_wait_tensorcnt(0);
  __syncthreads();

  for (int i = tid; i < 8192; i += THREADS) {           // expand: zero-pad, im=0
    const float v = (i < 8190) ? IM[i] : 0.0f;
    RE[i] = v;
    IM[i] = 0.0f;
  }
  __syncthreads();

  stage16(RE, IM, tid, +1, 1.0f);
  __syncthreads();
  rowfft(RE, IM, WR, WI, tid, +1);
  __syncthreads();

  float* dst = ws + (size_t)c * 16384;                  // coalesced planar store
  for (int i = tid; i < 8192; i += THREADS) {
    dst[i]        = RE[i];
    dst[i + 8192] = IM[i];
  }
}

// ---- kernel 2: per (b,c) fused fwd -> pointwise -> inv -> TDM scatter ------
__global__ void __launch_bounds__(THREADS)
tno_conv_kernel(const float* __restrict__ x, const float* __restrict__ tf,
                float* __restrict__ out) {
  extern __shared__ float lds[];
  float* RE = lds;
  float* IM = lds + 8192;
  float* WR = lds + 16384;
  float* WI = lds + 16640;
  const int tid = threadIdx.x;
  const int c = blockIdx.x;
  const int b = blockIdx.y;
  const size_t base = (size_t)b * 4096 * 512 + (size_t)c;

  if (tid < 32) tdm_load_col(IM, x + base, 4096u, 512u);
  {
    float s, co;
    __sincosf(-TWO_PI * (float)tid / 512.0f, &s, &co);
    WR[tid] = co; WI[tid] = s;
  }
  if (tid < 32) __builtin_amdgcn_s_wait_tensorcnt(0);
  __syncthreads();

  for (int i = tid; i < 8192; i += THREADS) {
    const float v = (i < 4096) ? IM[i] : 0.0f;
    RE[i] = v;
    IM[i] = 0.0f;
  }
  __syncthreads();

  stage16(RE, IM, tid, +1, 1.0f);                       // forward
  __syncthreads();
  rowfft(RE, IM, WR, WI, tid, +1);
  __syncthreads();

  const float* trp = tf + (size_t)c * 16384;            // pointwise (same order)
  const float* tip = trp + 8192;
  for (int i = tid; i < 8192; i += THREADS) {
    const float xr = RE[i], xi = IM[i];
    const float tr = trp[i], ti = tip[i];
    RE[i] = xr * tr - xi * ti;
    IM[i] = xr * ti + xi * tr;
  }
  __syncthreads();

  rowfft(RE, IM, WR, WI, tid, -1);                      // inverse
  __syncthreads();
  stage16(RE, IM, tid, -1, 1.0f / 8192.0f);
  __syncthreads();

  if (tid < 32) {                                        // out[b, 0..4095, c]
    tdm_store_col(RE, out + base, 4096u, 512u);
    __builtin_amdgcn_s_wait_tensorcnt(0);
  }
}

extern "C" void kernel_launch(void* const* d_in, const int* in_sizes, int n_in,
                              void* d_out, int out_size, void* d_ws, size_t ws_size,
                              hipStream_t stream) {
  (void)in_sizes; (void)n_in; (void)out_size; (void)ws_size;
  const float* x = (const float*)d_in[0];   // (4, 4096, 512) fp32
  const float* t = (const float*)d_in[1];   // (1, 8190, 512) fp32
  float* tf  = (float*)d_ws;                // 512 * 16384 floats = 32 MiB
  float* out = (float*)d_out;               // (4, 4096, 512) fp32
  const size_t shmem = (size_t)LDS_FLOATS * sizeof(float);

  tno_tfft_kernel<<<dim3(512), dim3(THREADS), shmem, stream>>>(t, tf);
  tno_conv_kernel<<<dim3(512, 4), dim3(THREADS), shmem, stream>>>(x, tf, out);
}